// BatchAllTripletLoss_40114994544729
// MI455X (gfx1250) — compile-verified
//
#include <hip/hip_runtime.h>
#include <hip/hip_bf16.h>

// ---------------------------------------------------------------------------
// BatchAllTripletLoss for MI455X (gfx1250, wave32).
//   B = 8192 anchors, D = 256 dims, 512 classes, MARGIN = 1.0, BIG = 1e9.
// Pipeline:
//   1) normalize_k : e = emb / max(||emb||, 1e-12)                (HBM read 8MB)
//   2) gram_min_k  : fused fp32 WMMA  G-tile -> clamp -> label-mask
//                    -> hardest-negative row min (never stores G)
//   3) loss_k      : per-anchor positive-pair losses (deterministic order)
//   4) finalize_k  : deterministic reduction -> scalar mean
// ---------------------------------------------------------------------------

#define B_N   8192
#define D_K   256

typedef __attribute__((ext_vector_type(2))) float v2f;
typedef __attribute__((ext_vector_type(8))) float v8f;

// --------------------------- 1) row L2-normalize ---------------------------
__global__ __launch_bounds__(256) void normalize_k(const float* __restrict__ emb,
                                                   float* __restrict__ e) {
    __shared__ float red[256];
    const int i = blockIdx.x;
    const int t = threadIdx.x;
    float v = emb[(size_t)i * D_K + t];
    red[t] = v * v;
    __syncthreads();
    #pragma unroll
    for (int s = 128; s > 0; s >>= 1) {
        if (t < s) red[t] += red[t + s];
        __syncthreads();
    }
    const float norm  = sqrtf(red[0]);
    const float scale = 1.0f / fmaxf(norm, 1e-12f);
    e[(size_t)i * D_K + t] = v * scale;
}

// ------------------------ init hardest-negative bits -----------------------
__global__ void init_k(unsigned* __restrict__ hb) {
    const int idx = blockIdx.x * 256 + threadIdx.x;
    if (idx < B_N) hb[idx] = __float_as_uint(1e9f);   // BIG, non-negative
}

// ------------- 2) fused Gram tile (fp32 WMMA) + hardest-neg min ------------
// One wave per workgroup; 64x64 output tile = 4x4 register-blocked 16x16 WMMA.
// Symmetric matrix: only bx >= by tiles computed; epilogue updates both the
// row-side mins (over cols) and col-side mins (over rows).
__global__ __launch_bounds__(32) void gram_min_k(const float* __restrict__ e,
                                                 const int*   __restrict__ labels,
                                                 unsigned*    __restrict__ hb) {
    const int bx = blockIdx.x, by = blockIdx.y;
    if (bx < by) return;                 // uniform per wave: EXEC stays all-ones
    const int lane = threadIdx.x;
    const int m = lane & 15;             // M (A) / N (B) index within 16
    const int h = lane >> 4;             // half: selects K pair {0,1} vs {2,3}
    const int row0 = by * 64, col0 = bx * 64;

    // Per-lane base pointers for the fp32 16x4 A/B fragment layout:
    // VGPR0 = K(2h), VGPR1 = K(2h+1)  -> contiguous float2 from row-major e.
    const float* ap[4]; const float* bp[4];
    #pragma unroll
    for (int t = 0; t < 4; ++t) {
        ap[t] = e + (size_t)(row0 + t * 16 + m) * D_K + 2 * h;
        bp[t] = e + (size_t)(col0 + t * 16 + m) * D_K + 2 * h;
    }

    v8f C[4][4];
    const v8f vz = {0.f, 0.f, 0.f, 0.f, 0.f, 0.f, 0.f, 0.f};
    #pragma unroll
    for (int a = 0; a < 4; ++a)
        #pragma unroll
        for (int b = 0; b < 4; ++b) C[a][b] = vz;

    // K loop: 64 steps of K=4; 8 loads feed 16 v_wmma_f32_16x16x4_f32.
    for (int kb = 0; kb < D_K; kb += 4) {
        v2f av[4], bv[4];
        #pragma unroll
        for (int t = 0; t < 4; ++t) av[t] = *(const v2f*)(ap[t] + kb);
        #pragma unroll
        for (int t = 0; t < 4; ++t) bv[t] = *(const v2f*)(bp[t] + kb);
        #pragma unroll
        for (int a = 0; a < 4; ++a)
            #pragma unroll
            for (int b = 0; b < 4; ++b)
                C[a][b] = __builtin_amdgcn_wmma_f32_16x16x4_f32(
                    false, av[a], false, bv[b], (short)0, C[a][b], false, false);
    }

    // Epilogue: d = max(0, 1 - dot); mask same-label; reduce mins.
    int lc[4];
    #pragma unroll
    for (int b = 0; b < 4; ++b) lc[b] = labels[col0 + b * 16 + m];

    float cmin[4] = {1e9f, 1e9f, 1e9f, 1e9f};
    #pragma unroll
    for (int a = 0; a < 4; ++a) {
        #pragma unroll
        for (int r = 0; r < 8; ++r) {
            const int   row = row0 + a * 16 + 8 * h + r;   // C layout: M=r+8h
            const int   lr  = labels[row];
            float rmin = 1e9f;
            #pragma unroll
            for (int b = 0; b < 4; ++b) {
                const float d = fmaxf(0.0f, 1.0f - C[a][b][r]);
                if (lr != lc[b]) {
                    rmin    = fminf(rmin, d);
                    cmin[b] = fminf(cmin[b], d);
                }
            }
            // reduce over the 16 lanes (columns) of this half
            rmin = fminf(rmin, __shfl_xor(rmin, 8, 16));
            rmin = fminf(rmin, __shfl_xor(rmin, 4, 16));
            rmin = fminf(rmin, __shfl_xor(rmin, 2, 16));
            rmin = fminf(rmin, __shfl_xor(rmin, 1, 16));
            // uint-bit atomicMin is a valid total order for non-negative floats
            if (m == 0 && rmin < 1e9f)
                atomicMin(hb + row, __float_as_uint(rmin));
        }
    }
    // column-side mins: each lane already holds min over its 32 rows; combine halves
    #pragma unroll
    for (int b = 0; b < 4; ++b) {
        const float cm = fminf(cmin[b], __shfl_xor(cmin[b], 16, 32));
        if (h == 0 && cm < 1e9f)
            atomicMin(hb + col0 + b * 16 + m, __float_as_uint(cm));
    }
}

// -------- 3) positive-pair losses, one block per anchor (deterministic) ----
__global__ __launch_bounds__(128) void loss_k(const float*    __restrict__ e,
                                              const int*      __restrict__ labels,
                                              const unsigned* __restrict__ hb,
                                              float*          __restrict__ psum,
                                              unsigned*       __restrict__ pcnt) {
    __shared__ float ei[D_K];
    __shared__ float sred[128];
    __shared__ unsigned cred[128];
    const int i = blockIdx.x;
    const int t = threadIdx.x;
    ei[t]       = e[(size_t)i * D_K + t];
    ei[t + 128] = e[(size_t)i * D_K + 128 + t];
    __syncthreads();

    const int   lab = labels[i];
    const float hn  = __uint_as_float(hb[i]);

    float    ls = 0.0f;
    unsigned lc = 0u;
    // fixed per-thread stride order -> bitwise-deterministic accumulation
    for (int j = t; j < B_N; j += 128) {
        if (j == i) continue;
        if (labels[j] != lab) continue;
        const float* ej = e + (size_t)j * D_K;
        float dot = 0.0f;
        #pragma unroll 8
        for (int k = 0; k < D_K; ++k) dot += ei[k] * ej[k];
        const float d = fmaxf(0.0f, 1.0f - dot);
        const float l = d - hn + 1.0f;          // MARGIN = 1.0
        if (l > 0.0f) { ls += l; lc += 1u; }
    }
    sred[t] = ls; cred[t] = lc;
    __syncthreads();
    #pragma unroll
    for (int s = 64; s > 0; s >>= 1) {
        if (t < s) { sred[t] += sred[t + s]; cred[t] += cred[t + s]; }
        __syncthreads();
    }
    if (t == 0) { psum[i] = sred[0]; pcnt[i] = cred[0]; }
}

// ----------------------- 4) deterministic final mean -----------------------
__global__ __launch_bounds__(256) void finalize_k(const float*    __restrict__ psum,
                                                  const unsigned* __restrict__ pcnt,
                                                  float*          __restrict__ out) {
    __shared__ float    s[256];
    __shared__ unsigned c[256];
    const int t = threadIdx.x;
    float    fs = 0.0f;
    unsigned fc = 0u;
    for (int j = t; j < B_N; j += 256) { fs += psum[j]; fc += pcnt[j]; }
    s[t] = fs; c[t] = fc;
    __syncthreads();
    #pragma unroll
    for (int st = 128; st > 0; st >>= 1) {
        if (t < st) { s[t] += s[t + st]; c[t] += c[t + st]; }
        __syncthreads();
    }
    if (t == 0) out[0] = (c[0] > 0u) ? (s[0] / (float)c[0]) : 0.0f;
}

// ---------------------------------------------------------------------------
extern "C" void kernel_launch(void* const* d_in, const int* in_sizes, int n_in,
                              void* d_out, int out_size, void* d_ws, size_t ws_size,
                              hipStream_t stream) {
    const float* emb    = (const float*)d_in[0];
    const int*   labels = (const int*)d_in[1];   // harness stages integers as int32
    float*       out    = (float*)d_out;

    // Workspace layout (~8.5 MB)
    float*    e    = (float*)d_ws;                         // B*D fp32 (8 MB)
    unsigned* hb   = (unsigned*)(e + (size_t)B_N * D_K);   // B   hardest-neg bits
    float*    psum = (float*)(hb + B_N);                   // B   per-anchor loss sums
    unsigned* pcnt = (unsigned*)(psum + B_N);              // B   per-anchor counts

    normalize_k<<<B_N, 256, 0, stream>>>(emb, e);
    init_k<<<B_N / 256, 256, 0, stream>>>(hb);
    gram_min_k<<<dim3(B_N / 64, B_N / 64), 32, 0, stream>>>(e, labels, hb);
    loss_k<<<B_N, 128, 0, stream>>>(e, labels, hb, psum, pcnt);
    finalize_k<<<1, 256, 0, stream>>>(psum, pcnt, out);
}